// SmoothnessLoss_7834020347941
// MI455X (gfx1250) — compile-verified
//
#include <hip/hip_runtime.h>
#include <stdint.h>

// SmoothnessLoss for MI455X (gfx1250, wave32).
// - Async global->LDS staging of 28x28x3x2 fp32 halos (ASYNCcnt path).
// - Branch-free fp32 VALU/TRANS stencil: all >0 masks are v_cndmask selects;
//   t^0.8 via raw v_log_f32/v_exp_f32 (t==0 -> log2=-inf -> exp2=+0, so no
//   guard branch is needed).
// - Cross-lane reductions via V_WMMA_F32_16X16X4_F32 with a ones operand
//   (f32 matrix pipe, no precision loss), deterministic two-pass finish.

typedef __attribute__((ext_vector_type(2))) float v2f;
typedef __attribute__((ext_vector_type(8))) float v8f;

#define TILE 16
#define PADR 6
#define HALO 28           // TILE + 2*PADR
#define LSTR 29           // +1 to avoid LDS bank conflicts
#define NTX 14            // 224/16
#define IMH 224
#define IMW 224
#define NBLK (NTX * NTX)  // 196

// Constants pre-folded into log2 domain (v_exp_f32 computes 2^x):
//   SIG_S = 1/(2*7*7),  SIG_S*log2(e)  = 0.014721377968254729
//   SIG_C = 1/(2*0.1^2) = 50, 50*log2(e) = 72.134752044448170
#define SIG_S_LOG2E 0.014721377968254729f
#define SIG_C_LOG2E 72.134752044448170f

__device__ __forceinline__ float fast_exp2(float x) {
  return __builtin_amdgcn_exp2f(x);   // v_exp_f32
}
__device__ __forceinline__ float fast_log2(float x) {
  return __builtin_amdgcn_logf(x);    // v_log_f32 (base-2)
}

// Sum of 32 lane partials via the matrix pipe:
// A = partials (A[m,0]=p[lane m], A[m,2]=p[lane m+16], A[m,1]=A[m,3]=0),
// B = ones  =>  D[m,n] = rowsum(m) = p[m] + p[m+16].
// Lanes 0-15 hold rows 0..7 in d[0..7], lanes 16-31 hold rows 8..15,
// so wave total = s8@lane0 + s8@lane16.
__device__ __forceinline__ float wave_wmma_reduce(float p) {
  v2f a; a[0] = p;    a[1] = 0.0f;
  v2f b; b[0] = 1.0f; b[1] = 1.0f;
  v8f d = {};
  d = __builtin_amdgcn_wmma_f32_16x16x4_f32(false, a, false, b,
                                            (short)0, d, false, false);
  return d[0] + d[1] + d[2] + d[3] + d[4] + d[5] + d[6] + d[7];
}

__global__ __launch_bounds__(256)
void loss_tile_kernel(const float* __restrict__ orig,
                      const float* __restrict__ smooth,
                      float* __restrict__ partials) {
  __shared__ float Sh[3 * HALO * LSTR];
  __shared__ float Oh[3 * HALO * LSTR];
  __shared__ float red[16];

  const int tid   = threadIdx.x;
  const int tileX = blockIdx.x % NTX;
  const int tileY = blockIdx.x / NTX;
  const int gx0   = tileX * TILE - PADR;
  const int gy0   = tileY * TILE - PADR;

  // ---- Stage halo tiles: async global->LDS for in-range, ds_store zeros for
  // out-of-range (disjoint addresses, so no ordering hazard between the two).
  for (int i = tid; i < 3 * HALO * HALO; i += 256) {
    int c   = i / (HALO * HALO);
    int r   = i - c * (HALO * HALO);
    int ly  = r / HALO;
    int lxh = r - ly * HALO;
    int gy  = gy0 + ly;
    int gx  = gx0 + lxh;
    int lidx = (c * HALO + ly) * LSTR + lxh;
    if (gy >= 0 && gy < IMH && gx >= 0 && gx < IMW) {
      size_t goff = ((size_t)c * IMH + gy) * IMW + gx;
      unsigned lds_s = (unsigned)(size_t)&Sh[lidx]; // low 32b of generic = LDS offset
      unsigned lds_o = (unsigned)(size_t)&Oh[lidx];
      uint64_t ga_s = (uint64_t)(smooth + goff);
      uint64_t ga_o = (uint64_t)(orig + goff);
      asm volatile("global_load_async_to_lds_b32 %0, %1, off"
                   :: "v"(lds_s), "v"(ga_s) : "memory");
      asm volatile("global_load_async_to_lds_b32 %0, %1, off"
                   :: "v"(lds_o), "v"(ga_o) : "memory");
    } else {
      Sh[lidx] = 0.0f;
      Oh[lidx] = 0.0f;
    }
  }
  asm volatile("s_wait_asynccnt 0" ::: "memory"); // drain this wave's ASYNCcnt
  __syncthreads();

  // ---- Per-pixel stencil from LDS.
  const int lx = tid & 15;
  const int ly = tid >> 4;
  const int cx = lx + PADR;
  const int cy = ly + PADR;
  const int gx = tileX * TILE + lx;
  const int gy = tileY * TILE + ly;

  float sc[3], oc[3];
#pragma unroll
  for (int c = 0; c < 3; ++c) {
    sc[c] = Sh[(c * HALO + cy) * LSTR + cx];
    oc[c] = Oh[(c * HALO + cy) * LSTR + cx];
  }

  // Edge responses: forward differences, zero at far border (branch-free:
  // border masks are uniform-ish selects, evaluated once per pixel).
  const float mx = (gx < IMW - 1) ? 1.0f : 0.0f;
  const float my = (gy < IMH - 1) ? 1.0f : 0.0f;
  float ero = 0.0f, ers = 0.0f;
#pragma unroll
  for (int c = 0; c < 3; ++c) {
    ero += mx * fabsf(Oh[(c * HALO + cy) * LSTR + cx + 1] - oc[c]);
    ers += mx * fabsf(Sh[(c * HALO + cy) * LSTR + cx + 1] - sc[c]);
    ero += my * fabsf(Oh[(c * HALO + cy + 1) * LSTR + cx] - oc[c]);
    ers += my * fabsf(Sh[(c * HALO + cy + 1) * LSTR + cx] - sc[c]);
  }
  const float mL = (ero < 20.0f && (ers - ero) > 10.0f) ? 1.0f : 0.0f;
  const float mS = 1.0f - mL;

  float acc = 0.0f;
  for (int oy = -PADR; oy <= PADR; ++oy) {
    // Row factor of the separable spatial weight, hoisted out of the ox loop.
    const float wsy = fast_exp2(-SIG_S_LOG2E * (float)(oy * oy));
    for (int ox = -PADR; ox <= PADR; ++ox) {
      float ws = wsy * fast_exp2(-SIG_S_LOG2E * (float)(ox * ox));
      float cd = 0.0f, t2 = 0.0f, tlp = 0.0f;
#pragma unroll
      for (int c = 0; c < 3; ++c) {
        int hidx = (c * HALO + cy + oy) * LSTR + (cx + ox);
        float Po = Oh[hidx];
        float Ps = Sh[hidx];
        // Masks as selects (v_cndmask), no exec-mask branches in the hot loop.
        float dp = (Po > 0.0f) ? (Po - oc[c]) : 0.0f;
        cd += dp * dp;
        float t  = (Ps > 0.0f) ? fabsf(Ps - sc[c]) : 0.0f;
        t2 += t * t;
        // t^0.8 = 2^(0.8*log2 t); at t==0: v_log_f32 -> -inf, v_exp_f32 -> +0,
        // which is exactly the reference's masked value, so no guard needed.
        tlp += fast_exp2(0.8f * fast_log2(t));
      }
      float wr = fast_exp2(-SIG_C_LOG2E * cd); // exp(-50*cd), cd >= 0
      acc += mL * ws * t2 + mS * wr * tlp;
    }
  }

  // ---- Block reduction: WMMA per wave, tiny LDS combine (deterministic).
  float s8 = wave_wmma_reduce(acc); // EXEC all-ones here (no divergence)
  int lane = tid & 31, wv = tid >> 5;
  if ((lane & 15) == 0)                 // lanes 0 and 16: single predicate
    red[2 * wv + (lane >> 4)] = s8;
  __syncthreads();
  if (tid == 0) {
    float tot = 0.0f;
#pragma unroll
    for (int i = 0; i < 16; ++i) tot += red[i];
    partials[blockIdx.x] = tot;
  }
}

__global__ __launch_bounds__(256)
void loss_finalize_kernel(const float* __restrict__ partials,
                          float* __restrict__ out) {
  __shared__ float red[16];
  const int tid = threadIdx.x;
  float p = (tid < NBLK) ? partials[tid] : 0.0f;
  float s8 = wave_wmma_reduce(p);
  int lane = tid & 31, wv = tid >> 5;
  if ((lane & 15) == 0)
    red[2 * wv + (lane >> 4)] = s8;
  __syncthreads();
  if (tid == 0) {
    float tot = 0.0f;
#pragma unroll
    for (int i = 0; i < 16; ++i) tot += red[i];
    out[0] = tot * (1.0f / (float)(IMH * IMW)); // / (n_pixels * B)
  }
}

extern "C" void kernel_launch(void* const* d_in, const int* in_sizes, int n_in,
                              void* d_out, int out_size, void* d_ws, size_t ws_size,
                              hipStream_t stream) {
  const float* orig   = (const float*)d_in[0]; // original_images (1,3,224,224)
  const float* smooth = (const float*)d_in[1]; // smooth_images   (1,3,224,224)
  float* partials = (float*)d_ws;              // 196 floats of scratch
  float* out = (float*)d_out;

  loss_tile_kernel<<<NBLK, 256, 0, stream>>>(orig, smooth, partials);
  loss_finalize_kernel<<<1, 256, 0, stream>>>(partials, out);
}